// KVCache_11682311045861
// MI455X (gfx1250) — compile-verified
//
#include <hip/hip_runtime.h>
#include <hip/hip_bf16.h>
#include <stdint.h>

// Problem dims from the reference (fixed by the harness).
#define B_ 4
#define H_ 32
#define S_ 8192
#define D_ 128
#define T_ 1024

typedef uint32_t v4u __attribute__((ext_vector_type(4)));   // 16B chunk
typedef uint32_t s4u __attribute__((ext_vector_type(4)));   // D# group0 (4 SGPRs)
typedef uint32_t s8u __attribute__((ext_vector_type(8)));   // D# group1 (8 SGPRs)

// ---------------- CDNA5 async global<->LDS helpers (ASYNCcnt path) ----------
__device__ __forceinline__ void async_load_b128(uint32_t lds_off, const void* g) {
    asm volatile("global_load_async_to_lds_b128 %0, %1, off th:TH_LOAD_NT"
                 :
                 : "v"(lds_off), "v"((uint64_t)(uintptr_t)g)
                 : "memory");
}
__device__ __forceinline__ void async_store_b128(void* g, uint32_t lds_off) {
    asm volatile("global_store_async_from_lds_b128 %0, %1, off th:TH_STORE_NT"
                 :
                 : "v"((uint64_t)(uintptr_t)g), "v"(lds_off)
                 : "memory");
}
__device__ __forceinline__ void wait_async0() {
    asm volatile("s_wait_asynccnt 0x0" ::: "memory");
}

// ---------------- CDNA5 Tensor Data Mover helpers (TENSORcnt path) ----------
// D# group 0 (128b): [1:0]=count=1, [63:32]=lds_addr, [120:64]=global_addr,
//                    [127:126]=type=2.
__device__ __forceinline__ s4u tdm_g0(uint32_t lds_addr, uint64_t gaddr) {
    s4u g0;
    g0[0] = 1u;                                         // count=1, user mode
    g0[1] = lds_addr;                                   // LDS byte address
    g0[2] = (uint32_t)gaddr;                            // global_addr[31:0]
    g0[3] = ((uint32_t)(gaddr >> 32) & 0x01FFFFFFu)     // global_addr[56:32]
          | (2u << 30);                                 // type = 2 ("image")
    return g0;
}
// D# group 1 (256b): 1-D tile of `nelem` 8-byte elements.
__device__ __forceinline__ s8u tdm_g1_1d(uint32_t nelem) {
    s8u g;
    g[0] = (3u << 16);                                  // data_size = 8 bytes
    g[1] = (nelem & 0xFFFFu) << 16;                     // tensor_dim0[15:0]
    g[2] = ((nelem >> 16) & 0xFFFFu) | (1u << 16);      // tensor_dim0[31:16], tensor_dim1=1
    g[3] = (nelem & 0xFFFFu) << 16;                     // tile_dim0
    g[4] = 1u;                                          // tile_dim1=1, tile_dim2=0
    g[5] = nelem;                                       // tensor_dim0_stride[31:0]
    g[6] = 0u;
    g[7] = 0u;
    return g;
}
__device__ __forceinline__ void tdm_load(const s4u& g0, const s8u& g1) {
    asm volatile("tensor_load_to_lds %0, %1" :: "s"(g0), "s"(g1) : "memory");
}
__device__ __forceinline__ void tdm_store(const s4u& g0, const s8u& g1) {
    asm volatile("tensor_store_from_lds %0, %1" :: "s"(g0), "s"(g1) : "memory");
}
__device__ __forceinline__ void wait_tensor(int n) {
    if (n == 0)      asm volatile("s_wait_tensorcnt 0x0" ::: "memory");
    else if (n == 1) asm volatile("s_wait_tensorcnt 0x1" ::: "memory");
    else             asm volatile("s_wait_tensorcnt 0x2" ::: "memory");
}

// ---------------------------------------------------------------------------
// TDM bulk copy, software-pipelined: one wave per block moves NTILES x 32 KB
// contiguous tiles through 3 LDS buffers. Tensor ops from one wave complete
// strictly in issue order, so s_wait_tensorcnt N == "all but last N done":
//   issue order: L0 L1 [S0 L2] [S1 L3] ... -> waits of <=2 are precise for
//   both the RAW (load->store) and WAR (store->reload buffer) hazards.
// ---------------------------------------------------------------------------
#define TDM_CHUNK   32768u   // bytes per tile
#define TDM_NTILES  8        // tiles per block
__global__ __launch_bounds__(32) void
copy_cache_tdm(const uint8_t* __restrict__ src, uint8_t* __restrict__ dst) {
    __shared__ __align__(16) uint8_t buf[3][TDM_CHUNK];  // 96 KB staging
    const uint32_t lds0  = (uint32_t)(uintptr_t)&buf[0][0];
    const uint32_t nelem = TDM_CHUNK >> 3;               // 8-byte elements
    const uint64_t base  = (uint64_t)blockIdx.x * (TDM_CHUNK * TDM_NTILES)
                         + (uint64_t)(uintptr_t)src;
    const uint64_t dbase = (uint64_t)blockIdx.x * (TDM_CHUNK * TDM_NTILES)
                         + (uint64_t)(uintptr_t)dst;
    const s8u g1 = tdm_g1_1d(nelem);

    // Prime: loads of tiles 0 and 1 into buffers 0 and 1.
    tdm_load(tdm_g0(lds0 + 0u * TDM_CHUNK, base + 0ull * TDM_CHUNK), g1);
    tdm_load(tdm_g0(lds0 + 1u * TDM_CHUNK, base + 1ull * TDM_CHUNK), g1);

#pragma unroll
    for (int t = 0; t < TDM_NTILES; ++t) {
        const uint32_t slot = (uint32_t)(t % 3) * TDM_CHUNK;
        wait_tensor(t == 0 ? 1 : 2);                 // L(t) complete
        tdm_store(tdm_g0(lds0 + slot, dbase + (uint64_t)t * TDM_CHUNK), g1);
        if (t + 2 < TDM_NTILES) {
            wait_tensor(2);                          // S(t-1) complete -> buffer free
            const uint32_t nslot = (uint32_t)((t + 2) % 3) * TDM_CHUNK;
            tdm_load(tdm_g0(lds0 + nslot, base + (uint64_t)(t + 2) * TDM_CHUNK), g1);
        }
    }
    wait_tensor(0);                                  // drain before endpgm
}

// ---------------------------------------------------------------------------
// Async-LDS bulk copy (ASYNCcnt path): each thread streams 16B chunks through
// a private LDS slot; load(i+1) overlaps store(i) in flight.
// ---------------------------------------------------------------------------
__global__ __launch_bounds__(256) void
copy_cache_async(const v4u* __restrict__ src, v4u* __restrict__ dst,
                 unsigned long long n_chunks, int iters) {
    __shared__ v4u buf[256];
    const uint32_t lds_off = (uint32_t)(uintptr_t)(&buf[threadIdx.x]);
    const size_t stride = (size_t)gridDim.x * blockDim.x;
    size_t i = (size_t)blockIdx.x * blockDim.x + threadIdx.x;

    for (int it = 0; it < iters; ++it, i += stride) {
        if (i < n_chunks) {
            async_load_b128(lds_off, src + i);
            wait_async0();               // drains this load + previous store
            async_store_b128(dst + i, lds_off);
        }
    }
    wait_async0();
}

// ---------------------------------------------------------------------------
// Scatter: out_k[b, h, pos[b,t], :] = k_val[b, h, t, :]   (same for V)
// idx bits (all dims pow2): c[3:0] t[13:4] h[18:14] b[20:19]
// ---------------------------------------------------------------------------
__global__ __launch_bounds__(256) void
scatter_vals(const v4u* __restrict__ kval, const v4u* __restrict__ vval,
             const int* __restrict__ pos,
             v4u* __restrict__ kout, v4u* __restrict__ vout) {
    const size_t idx = (size_t)blockIdx.x * blockDim.x + threadIdx.x;
    const uint32_t c = (uint32_t)idx & 15u;
    const uint32_t t = ((uint32_t)idx >> 4) & (T_ - 1);
    const uint32_t h = ((uint32_t)idx >> 14) & (H_ - 1);
    const uint32_t b = (uint32_t)idx >> 19;

    const int p = pos[b * T_ + t];
    const size_t dsti = ((size_t)(b * H_ + h) * S_ + (size_t)p) * 16u + c;

    const v4u kchunk = __builtin_nontemporal_load(&kval[idx]);
    const v4u vchunk = __builtin_nontemporal_load(&vval[idx]);
    __builtin_nontemporal_store(kchunk, &kout[dsti]);
    __builtin_nontemporal_store(vchunk, &vout[dsti]);
}

// ---------------------------------------------------------------------------
extern "C" void kernel_launch(void* const* d_in, const int* in_sizes, int n_in,
                              void* d_out, int out_size, void* d_ws, size_t ws_size,
                              hipStream_t stream) {
    (void)in_sizes; (void)n_in; (void)out_size; (void)d_ws; (void)ws_size;

    const uint8_t* k_cache = (const uint8_t*)d_in[0];
    const v4u*     v_cache = (const v4u*)d_in[1];
    const v4u*     k_val   = (const v4u*)d_in[2];
    const v4u*     v_val   = (const v4u*)d_in[3];
    const int*     pos     = (const int*)d_in[4];

    const unsigned long long cache_bytes  =
        (unsigned long long)B_ * H_ * S_ * D_ * 2ull;      // 256 MiB per tensor
    const unsigned long long cache_chunks = cache_bytes / 16ull;
    const unsigned long long val_chunks   =
        (unsigned long long)B_ * H_ * T_ * D_ / 8ull;      // 2,097,152

    uint8_t* out_k = (uint8_t*)d_out;
    v4u*     out_v = (v4u*)d_out + cache_chunks;

    // Phase 1a: K cache via Tensor Data Mover (8 x 32 KB tiles per block,
    // 3-buffer pipelined; 256 MiB / 256 KiB = 1024 blocks).
    copy_cache_tdm<<<(unsigned)(cache_bytes / (TDM_CHUNK * TDM_NTILES)), 32, 0,
                     stream>>>(k_cache, out_k);

    // Phase 1b: V cache via async global<->LDS b128 streaming.
    const int ITERS = 8;
    copy_cache_async<<<(unsigned)(cache_chunks / (256ull * ITERS)), 256, 0,
                       stream>>>(v_cache, out_v, cache_chunks, ITERS);

    // Phase 2: scatter the new values into the copied caches.
    scatter_vals<<<(unsigned)(val_chunks / 256ull), 256, 0, stream>>>(
        k_val, v_val, pos, (v4u*)out_k, out_v);
}